// MetaGAT_34926674051560
// MI455X (gfx1250) — compile-verified
//
#include <hip/hip_runtime.h>

typedef __attribute__((ext_vector_type(2))) float v2f;
typedef __attribute__((ext_vector_type(8))) float v8f;

#define NODE_STRIDE 512   // B*H = 16*32
#define EPB 8             // edges per block (one wave32 per edge)
#define ROWP 68           // padded LDS row stride (16B-aligned, conflict-free)

#if defined(__has_builtin)
#  if __has_builtin(__builtin_amdgcn_global_load_async_to_lds_b128) && \
      __has_builtin(__builtin_amdgcn_s_wait_asynccnt)
#    define USE_ASYNC_LDS 1
#  endif
#endif
#ifndef USE_ASYNC_LDS
#  define USE_ASYNC_LDS 0
#endif

#if USE_ASYNC_LDS
// builtin expects: (v4i addrspace(1)*, v4i addrspace(3)*, imm offset, imm cpol)
typedef int v4i_vs __attribute__((vector_size(16)));
typedef v4i_vs __attribute__((address_space(1)))* g128p;
typedef v4i_vs __attribute__((address_space(3)))* l128p;
#endif

// fast sigmoid: v_mul + v_exp (TRANS) + v_add + v_rcp (TRANS), no IEEE div chain
__device__ __forceinline__ float fast_sigmoid(float x) {
    return __builtin_amdgcn_rcpf(1.0f + __expf(-x));
}

__global__ void init_kernel(float* __restrict__ den, float* __restrict__ out,
                            int total) {
    int i = blockIdx.x * blockDim.x + threadIdx.x;
    if (i < total) { den[i] = 0.0f; out[i] = 0.0f; }
}

__global__ __launch_bounds__(256) void edge_pass_kernel(
    const float* __restrict__ state, const float* __restrict__ feature,
    const float* __restrict__ dist,
    const float* __restrict__ w1, const float* __restrict__ b1,
    const float* __restrict__ w2, const float* __restrict__ b2,
    const float* __restrict__ w3, const float* __restrict__ b3,
    const int* __restrict__ src, const int* __restrict__ dst,
    float* __restrict__ den, float* __restrict__ num, int E)
{
    __shared__ float ef[EPB][96];          // per-edge MLP input
    __shared__ float h1s[EPB][16];         // hidden layer 1
    __shared__ float sbuf[EPB][16 * ROWP]; // s = concat(state[src],state[dst])

    const int wave = threadIdx.x >> 5;
    const int lane = threadIdx.x & 31;
    const int e = blockIdx.x * EPB + wave;
    const bool active = (e < E);           // wave-uniform

    int si = 0, di = 0;
    if (active) {
        si = src[e]; di = dst[e];
        const float* srow = state + (size_t)si * NODE_STRIDE;
        const float* drow = state + (size_t)di * NODE_STRIDE;
#if USE_ASYNC_LDS
        // stage s[16][64] via async global->LDS B128: 8 chunks of 4 floats/lane,
        // issued first so the copy overlaps the MLP feature gathers below
        {
            const int c0 = lane;           // chunk = row*16 + k/4
#pragma unroll
            for (int t = 0; t < 8; ++t) {
                const int c = c0 + 32 * t;
                const int row = c >> 4;
                const int k = (c & 15) * 4;
                const float* gp = (k < 32) ? (srow + row * 32 + k)
                                           : (drow + row * 32 + (k - 32));
                __builtin_amdgcn_global_load_async_to_lds_b128(
                    (g128p)gp, (l128p)&sbuf[wave][row * ROWP + k], 0, 0);
            }
        }
#else
        for (int idx = lane; idx < 16 * 64; idx += 32) {
            int row = idx >> 6, k = idx & 63;
            float v = (k < 32) ? srow[row * 32 + k] : drow[row * 32 + (k - 32)];
            sbuf[wave][row * ROWP + k] = v;
        }
#endif
        // efeat = [feature[src](47) | feature[dst](47) | dist(2)]
        for (int j = lane; j < 96; j += 32) {
            float v;
            if (j < 47)      v = feature[si * 47 + j];
            else if (j < 94) v = feature[di * 47 + (j - 47)];
            else             v = dist[e * 2 + (j - 94)];
            ef[wave][j] = v;
        }
    }
    __syncthreads();

    // layer 1: 96 -> 16, sigmoid
    if (active && lane < 16) {
        float acc = b1[lane];
        for (int j = 0; j < 96; ++j) acc += ef[wave][j] * w1[j * 16 + lane];
        h1s[wave][lane] = fast_sigmoid(acc);
    }
    __syncthreads();
    if (!active) return;                   // wave-uniform, no barriers below

    // layer 2: 16 -> 2, sigmoid (redundant per lane; uniform LDS reads)
    float h20 = b2[0], h21 = b2[1];
    for (int j = 0; j < 16; ++j) {
        float v = h1s[wave][j];
        h20 += v * w2[j * 2 + 0];
        h21 += v * w2[j * 2 + 1];
    }
    h20 = fast_sigmoid(h20);
    h21 = fast_sigmoid(h21);

#if USE_ASYNC_LDS
    __builtin_amdgcn_s_wait_asynccnt(0);   // s-tile resident in LDS from here
#endif

    // ---- alpha = s[16,64] @ W[64,32] via V_WMMA_F32_16X16X4_F32 ----
    // A 16x4 layout: lane L -> M = L&15; VGPR v holds K = 4*kk + 2*(L>>4) + v
    // B 4x16 layout: lane L -> N = L&15; VGPR v holds K = 4*kk + 2*(L>>4) + v
    // C/D 16x16:     lane L -> N = L&15; VGPR r holds M = r + 8*(L>>4)
    const int g    = lane >> 4;
    const int rowA = lane & 15;
    const int n0   = lane & 15;
    const float* w3a = w3;           // row 0 of w3 [2,2048]
    const float* w3b = w3 + 2048;    // row 1

    v8f c0 = {0.f, 0.f, 0.f, 0.f, 0.f, 0.f, 0.f, 0.f};
    v8f c1 = {0.f, 0.f, 0.f, 0.f, 0.f, 0.f, 0.f, 0.f};

#pragma unroll
    for (int kk = 0; kk < 16; ++kk) {
        const int kq0 = 4 * kk + 2 * g;
        const int kq1 = kq0 + 1;
        v2f a;
        a.x = sbuf[wave][rowA * ROWP + kq0];
        a.y = sbuf[wave][rowA * ROWP + kq1];
        // W[k][n] = sigmoid(h20*w3[0,k*32+n] + h21*w3[1,k*32+n] + b3[k*32+n])
        const int i00 = kq0 * 32 + n0;
        const int i10 = kq1 * 32 + n0;
        v2f b0, b1v;
        b0.x  = fast_sigmoid(fmaf(h20, w3a[i00],      fmaf(h21, w3b[i00],      b3[i00])));
        b0.y  = fast_sigmoid(fmaf(h20, w3a[i10],      fmaf(h21, w3b[i10],      b3[i10])));
        b1v.x = fast_sigmoid(fmaf(h20, w3a[i00 + 16], fmaf(h21, w3b[i00 + 16], b3[i00 + 16])));
        b1v.y = fast_sigmoid(fmaf(h20, w3a[i10 + 16], fmaf(h21, w3b[i10 + 16], b3[i10 + 16])));

        c0 = __builtin_amdgcn_wmma_f32_16x16x4_f32(false, a, false, b0,
                                                   (short)0, c0, false, false);
        c1 = __builtin_amdgcn_wmma_f32_16x16x4_f32(false, a, false, b1v,
                                                   (short)0, c1, false, false);
    }

    // epilogue: leaky_relu, shift-free softmax numerators (alpha provably << 60),
    // fused den/num scatter-reduce per dst
#pragma unroll
    for (int r = 0; r < 8; ++r) {
        const int m = r + 8 * g;
        float a0 = c0[r]; a0 = (a0 > 0.f) ? a0 : 0.01f * a0;
        float a1 = c1[r]; a1 = (a1 > 0.f) ? a1 : 0.01f * a1;
        const int base = di * NODE_STRIDE + m * 32;
        const int idx0 = base + n0;
        const int idx1 = base + n0 + 16;
        const float ex0 = __expf(fminf(a0, 60.f));
        const float ex1 = __expf(fminf(a1, 60.f));
        atomicAdd(&den[idx0], ex0);
        atomicAdd(&den[idx1], ex1);
        // state[src][m][n] is the k<32 half of the staged tile
        const float s0 = sbuf[wave][m * ROWP + n0];
        const float s1 = sbuf[wave][m * ROWP + n0 + 16];
        atomicAdd(&num[idx0], ex0 * s0);
        atomicAdd(&num[idx1], ex1 * s1);
    }
}

__global__ void finalize_kernel(float* __restrict__ out,
                                const float* __restrict__ den,
                                const float* __restrict__ weight, int total) {
    int i = blockIdx.x * blockDim.x + threadIdx.x;
    if (i < total) {
        const float d  = den[i];
        const float dg = (d > 0.f) ? d : 1.f;
        const float sw = __builtin_amdgcn_rcpf(1.f + __expf(-weight[0]));
        const float v  = out[i] * __builtin_amdgcn_rcpf(dg) * sw;
        out[i] = (v > 0.f) ? v : 0.f;
    }
}

extern "C" void kernel_launch(void* const* d_in, const int* in_sizes, int n_in,
                              void* d_out, int out_size, void* d_ws, size_t ws_size,
                              hipStream_t stream) {
    const float* state   = (const float*)d_in[0];
    const float* feature = (const float*)d_in[1];
    const float* dist    = (const float*)d_in[2];
    const float* w1      = (const float*)d_in[3];
    const float* b1      = (const float*)d_in[4];
    const float* w2      = (const float*)d_in[5];
    const float* b2      = (const float*)d_in[6];
    const float* w3      = (const float*)d_in[7];
    const float* b3      = (const float*)d_in[8];
    const float* weight  = (const float*)d_in[9];
    const int*   src     = (const int*)d_in[10];
    const int*   dst     = (const int*)d_in[11];

    const int E = in_sizes[10];
    const int total = out_size;           // N * B * H

    float* den = (float*)d_ws;
    float* out = (float*)d_out;

    const int tb = 256;
    init_kernel<<<(total + tb - 1) / tb, tb, 0, stream>>>(den, out, total);

    const int eblocks = (E + EPB - 1) / EPB;
    edge_pass_kernel<<<eblocks, tb, 0, stream>>>(
        state, feature, dist, w1, b1, w2, b2, w3, b3, src, dst,
        den, out, E);

    finalize_kernel<<<(total + tb - 1) / tb, tb, 0, stream>>>(out, den, weight, total);
}